// ArrivalTime_15891378995298
// MI455X (gfx1250) — compile-verified
//
#include <hip/hip_runtime.h>

#define B_  64
#define S_  512
#define D_  256
#define H_  4
#define HD_ 64
#define T_  24
#define N_  (B_ * S_)   // 32768
#define KDIM 96         // H_ * T_

typedef float v2f __attribute__((ext_vector_type(2)));
typedef float v8f __attribute__((ext_vector_type(8)));

// ---------------- prep1: k, v, time-query, user-query (tiny dot products) ----
__global__ void prep1(const float* __restrict__ ts,     // [24,256]
                      const int*   __restrict__ user,   // [64]
                      const float* __restrict__ upref,  // [NU,256]
                      const float* __restrict__ Wq, const float* __restrict__ bq,
                      const float* __restrict__ Wk, const float* __restrict__ bk,
                      const float* __restrict__ Wv, const float* __restrict__ bv,
                      float* __restrict__ Kd, float* __restrict__ Vd,
                      float* __restrict__ TQ, float* __restrict__ UQ)
{
    int idx = blockIdx.x * blockDim.x + threadIdx.x;
    if (idx < 6144) {                       // K: [h][t][c] = h*1536 + t*64 + c
        int h = idx / 1536, t = (idx % 1536) / 64, c = idx % 64;
        const float* w = Wk + (h * 64 + c) * 256;
        const float* x = ts + t * 256;
        float s = bk[h * 64 + c];
        for (int d = 0; d < 256; ++d) s += x[d] * w[d];
        Kd[idx] = s;
    } else if (idx < 12288) {               // V
        int j = idx - 6144;
        int h = j / 1536, t = (j % 1536) / 64, c = j % 64;
        const float* w = Wv + (h * 64 + c) * 256;
        const float* x = ts + t * 256;
        float s = bv[h * 64 + c];
        for (int d = 0; d < 256; ++d) s += x[d] * w[d];
        Vd[j] = s;
    } else if (idx < 18432) {               // TQ: [tau][hc] = tau*256 + hc
        int j = idx - 12288;
        int tau = j / 256, hc = j % 256;
        const float* w = Wq + hc * 512 + 256;     // time half of Wq
        const float* x = ts + tau * 256;
        float s = 0.f;
        for (int d = 0; d < 256; ++d) s += x[d] * w[d];
        TQ[j] = s;
    } else if (idx < 34816) {               // UQ: [b][hc] = b*256 + hc (incl. bq)
        int j = idx - 18432;
        int b = j / 256, hc = j % 256;
        const float* w = Wq + hc * 512;           // user half of Wq
        const float* x = upref + (size_t)user[b] * 256;
        float s = bq[hc];
        for (int d = 0; d < 256; ++d) s += x[d] * w[d];
        UQ[j] = s;
    }
}

// ---------------- prep2: ST, SU (scale folded), VW = v @ Wu-slice ------------
__global__ void prep2(const float* __restrict__ Kd, const float* __restrict__ Vd,
                      const float* __restrict__ TQ, const float* __restrict__ UQ,
                      const float* __restrict__ Wu,
                      float* __restrict__ ST, float* __restrict__ SU,
                      float* __restrict__ VW)
{
    const float scale = 0.125f;             // 1/sqrt(64)
    int idx = blockIdx.x * blockDim.x + threadIdx.x;
    if (idx < 2304) {                       // ST[tau*96 + h*24 + t]
        int tau = idx / 96, h = (idx % 96) / 24, t = idx % 24;
        const float* q = TQ + tau * 256 + h * 64;
        const float* k = Kd + (h * 24 + t) * 64;
        float s = 0.f;
        for (int c = 0; c < 64; ++c) s += q[c] * k[c];
        ST[idx] = s * scale;
    } else if (idx < 8448) {                // SU[b*96 + h*24 + t]
        int j = idx - 2304;
        int b = j / 96, h = (j % 96) / 24, t = j % 24;
        const float* q = UQ + b * 256 + h * 64;
        const float* k = Kd + (h * 24 + t) * 64;
        float s = 0.f;
        for (int c = 0; c < 64; ++c) s += q[c] * k[c];
        SU[j] = s * scale;
    } else if (idx < 33024) {               // VW[(h*24+t)*256 + d]
        int j = idx - 8448;
        int ht = j / 256, d = j % 256;
        int h = ht / 24;
        const float* v = Vd + ht * 64;
        const float* w = Wu + d * 256 + h * 64;
        float s = 0.f;
        for (int c = 0; c < 64; ++c) s += v[c] * w[c];
        VW[j] = s;
    }
}

// ---------------- main: softmax(24) + WMMA GEMM (Nx96)@(96x256) --------------
__global__ __launch_bounds__(128) void attn_main(
    const int*   __restrict__ hour,    // [N]
    const int*   __restrict__ hmask,   // [N,24]
    const float* __restrict__ SU,      // [64,96]
    const float* __restrict__ ST,      // [24,96]
    const float* __restrict__ VW,      // [96,256]
    const float* __restrict__ bu,      // [256]
    float*       __restrict__ out)     // [N,256]
{
    __shared__ float attnS[64][100];   // stride 100 dwords: conflict-free
    const int row0 = blockIdx.x * 64;
    const int tid  = threadIdx.x;

    // ---- phase 1: masked softmax, 2 threads per token (2 heads each)
    {
        int local = tid >> 1;
        int h0 = (tid & 1) * 2;
        int n = row0 + local;
        int b = n >> 9;                // n / 512
        int hr = hour[n];
        const int* mp = hmask + (size_t)n * 24;
        int m[24];
        #pragma unroll
        for (int t = 0; t < 24; ++t) m[t] = mp[t];
        for (int hh = 0; hh < 2; ++hh) {
            int h = h0 + hh;
            const float* sup = SU + b * 96 + h * 24;
            const float* stp = ST + hr * 96 + h * 24;
            float sc[24];
            float mx = -3.0e38f;
            #pragma unroll
            for (int t = 0; t < 24; ++t) {
                float x = (m[t] == 1) ? -3.0e38f : (sup[t] + stp[t]);
                sc[t] = x;
                mx = fmaxf(mx, x);
            }
            float sum = 0.f;
            #pragma unroll
            for (int t = 0; t < 24; ++t) {
                float e = (sc[t] <= -1.0e38f) ? 0.f : __expf(sc[t] - mx);
                sc[t] = e;
                sum += e;
            }
            float inv = (sum > 0.f) ? (1.f / sum) : 0.f;
            #pragma unroll
            for (int t = 0; t < 24; ++t) attnS[local][h * 24 + t] = sc[t] * inv;
        }
    }
    __syncthreads();

    // ---- phase 2: per wave, 16 rows x 256 cols via v_wmma_f32_16x16x4_f32
    const int wave = tid >> 5;
    const int lane = tid & 31;
    const int mrow = lane & 15;        // row / col within 16
    const int kh   = (lane >> 4) * 2;  // K sub-offset: lanes 16-31 hold K+2,K+3
    const int rbase = wave * 16;
    const float* arow = &attnS[rbase + mrow][0];

    for (int ct = 0; ct < 16; ++ct) {
        const int col = ct * 16 + mrow;
        v8f acc = {};
        #pragma unroll
        for (int k0 = 0; k0 < 96; k0 += 4) {
            v2f a, bm;
            a.x  = arow[k0 + kh];
            a.y  = arow[k0 + kh + 1];
            bm.x = VW[(k0 + kh) * 256 + col];
            bm.y = VW[(k0 + kh + 1) * 256 + col];
            acc = __builtin_amdgcn_wmma_f32_16x16x4_f32(
                      false, a, false, bm, (short)0, acc, false, false);
        }
        const float bub = bu[col];
        #pragma unroll
        for (int j = 0; j < 8; ++j) {  // C/D layout: VGPR j -> rows j, j+8
            int r = row0 + rbase + (lane >> 4) * 8 + j;
            out[(size_t)r * 256 + col] = acc[j] + bub;
        }
    }
}

extern "C" void kernel_launch(void* const* d_in, const int* in_sizes, int n_in,
                              void* d_out, int out_size, void* d_ws, size_t ws_size,
                              hipStream_t stream) {
    const float* ts    = (const float*)d_in[0];
    const int*   user  = (const int*)  d_in[1];
    const int*   hour  = (const int*)  d_in[2];
    const int*   hmask = (const int*)  d_in[3];
    const float* upref = (const float*)d_in[4];
    const float* Wq    = (const float*)d_in[5];
    const float* bq    = (const float*)d_in[6];
    const float* Wk    = (const float*)d_in[7];
    const float* bk    = (const float*)d_in[8];
    const float* Wv    = (const float*)d_in[9];
    const float* bv    = (const float*)d_in[10];
    const float* Wu    = (const float*)d_in[11];
    const float* bu    = (const float*)d_in[12];
    float* out = (float*)d_out;

    float* w  = (float*)d_ws;
    float* Kd = w;              // 6144
    float* Vd = Kd + 6144;      // 6144
    float* TQ = Vd + 6144;      // 6144
    float* UQ = TQ + 6144;      // 16384
    float* ST = UQ + 16384;     // 2304
    float* SU = ST + 2304;      // 6144
    float* VW = SU + 6144;      // 24576

    prep1<<<(34816 + 255) / 256, 256, 0, stream>>>(ts, user, upref, Wq, bq,
                                                   Wk, bk, Wv, bv, Kd, Vd, TQ, UQ);
    prep2<<<(33024 + 255) / 256, 256, 0, stream>>>(Kd, Vd, TQ, UQ, Wu, ST, SU, VW);
    attn_main<<<N_ / 64, 128, 0, stream>>>(hour, hmask, SU, ST, VW, bu, out);
}